// TemporalNeighborSampler_47364899340960
// MI455X (gfx1250) — compile-verified
//
#include <hip/hip_runtime.h>
#include <stdint.h>

// Problem constants (from the reference): MAX_DEG=128, NUM_SAMPLES=32.
#define MAX_DEG 128
#define NSAMP   32
#define WPB     8      // waves per block (256 threads)
#define QPW     4      // queries per wave (software-pipelined, double-buffered)

// One async DMA instruction moves 16B/lane = 512B/wave from global to LDS.
// VDST operand = LDS byte address (low 32 bits of the generic __shared__ ptr),
// VADDR = 64-bit global address, saddr off. Tracked by ASYNCcnt.
__device__ __forceinline__ void async_row_b128(uint32_t lds_addr, uint64_t gaddr) {
  asm volatile("global_load_async_to_lds_b128 %0, %1, off"
               :: "v"(lds_addr), "v"(gaddr) : "memory");
}

__global__ __launch_bounds__(WPB * 32)
void temporal_sampler(const int* __restrict__ adj, const float* __restrict__ ts,
                      const int* __restrict__ ids, const float* __restrict__ tss,
                      int num_ids, int* __restrict__ out_nb, float* __restrict__ out_ts)
{
  __shared__ float lds_ts[WPB][2][MAX_DEG];   // double-buffered ts row per wave
  const int lane  = threadIdx.x & 31;
  const int wave  = threadIdx.x >> 5;
  const int qbase = (blockIdx.x * WPB + wave) * QPW;

  int   node[2];
  float thr[2];
  int   qidx[2];
  bool  qok[2];

  auto issue = [&](int q, int slot) {
    const bool ok = (q < num_ids);
    // Explicitly uniform query index -> scalar s_load for ids/tss.
    const int qc = __builtin_amdgcn_readfirstlane(ok ? q : (num_ids - 1));
    const int   nd = ids[qc];
    const float t  = tss[qc];
    node[slot] = nd; thr[slot] = t; qidx[slot] = qc; qok[slot] = ok;
    // Async DMA the whole sorted ts row (512B) into this wave's LDS buffer.
    const uint64_t g = (uint64_t)(uintptr_t)(ts + (size_t)nd * MAX_DEG + lane * 4);
    const uint32_t l = (uint32_t)(uintptr_t)(const void*)&lds_ts[wave][slot][lane * 4];
    async_row_b128(l, g);
    // Window cols land in [count-32, count) with count in [~64,128] =>
    // bytes [128,512) of the adj row. All lanes prefetch (one instruction),
    // addresses clamped onto those 3 cachelines -- no exec divergence.
    int pf = lane & 3; pf = pf > 2 ? 2 : pf;                  // 0,1,2,2 pattern
    __builtin_prefetch(adj + (size_t)nd * MAX_DEG + 32 + pf * 32, 0, 0);
  };

  issue(qbase, 0);

  #pragma unroll
  for (int i = 0; i < QPW; ++i) {
    const int buf = i & 1;
    if (i + 1 < QPW) {
      issue(qbase + i + 1, (i + 1) & 1);
      // Async loads complete in order: <=1 outstanding means buffer i is ready.
      asm volatile("s_wait_asynccnt 0x1" ::: "memory");
    } else {
      asm volatile("s_wait_asynccnt 0x0" ::: "memory");
    }

    const float t  = thr[buf];
    const int   nd = node[buf];
    const int   q  = qidx[buf];

    // Rank of threshold t in the sorted row: 4x (LDS read + ballot + popc),
    // result wave-uniform, no shuffle tree needed on wave32.
    unsigned cnt = 0;
    #pragma unroll
    for (int c = 0; c < 4; ++c) {
      const float v = lds_ts[wave][buf][c * 32 + lane];
      cnt += (unsigned)__popcll(__ballot(v < t));
    }

    int col = (int)cnt - NSAMP + lane;
    col = col < 0 ? 0 : (col > MAX_DEG - 1 ? MAX_DEG - 1 : col);

    const float tsv = lds_ts[wave][buf][col];                 // conflict-free: 32 consecutive banks
    const int   nb  = adj[(size_t)nd * MAX_DEG + col];        // coalesced 128B gather (prefetched)

    if (qok[buf]) {
      out_nb[(size_t)q * NSAMP + lane] = nb;                  // 128B coalesced store
      out_ts[(size_t)q * NSAMP + lane] = tsv;                 // 128B coalesced store
    }
  }
}

extern "C" void kernel_launch(void* const* d_in, const int* in_sizes, int n_in,
                              void* d_out, int out_size, void* d_ws, size_t ws_size,
                              hipStream_t stream) {
  const int*   adj = (const int*)  d_in[0];   // [N_NODES, 128] int32
  const float* ts  = (const float*)d_in[1];   // [N_NODES, 128] float32 (row-sorted)
  const int*   ids = (const int*)  d_in[2];   // [num_ids] int32
  const float* tss = (const float*)d_in[3];   // [num_ids] float32
  const int num_ids = in_sizes[2];

  // d_out = [neighbors (int32 bits), tss_out (float32)] concatenated flat.
  int*   out_nb = (int*)d_out;
  float* out_ts = (float*)d_out + (size_t)num_ids * NSAMP;

  const int qpb    = WPB * QPW;               // 32 queries per block
  const int blocks = (num_ids + qpb - 1) / qpb;
  temporal_sampler<<<blocks, WPB * 32, 0, stream>>>(adj, ts, ids, tss,
                                                    num_ids, out_nb, out_ts);
}